// User_7206955122815
// MI455X (gfx1250) — compile-verified
//
#include <hip/hip_runtime.h>
#include <math.h>

typedef float v2f __attribute__((ext_vector_type(2)));
typedef float v8f __attribute__((ext_vector_type(8)));

#define BB 16384
#define LL 512
#define SUBSEG 128                     // positions per lane stream
#define NIT (SUBSEG / 2)               // 64 iterations, 2 positions/lane/iter
#define GROUPS_PER_BLOCK 2             // 16-row groups per block
#define ROWS_PER_BLOCK (GROUPS_PER_BLOCK * 16)   // 32 rows
#define THREADS (GROUPS_PER_BLOCK * 2 * 32)      // 2 segment-waves per group
#define PADV 3.0f

__global__ __launch_bounds__(THREADS)
void crowd_beta_kernel(const float* __restrict__ inp,
                       const float* __restrict__ w,
                       float* __restrict__ out)
{
    // per 16-row group: 4 sub-segment partials per row
    __shared__ float sR[GROUPS_PER_BLOCK][4][16];   // real partial sums
    __shared__ float sF[GROUPS_PER_BLOCK][4][16];   // fake partial sums
    __shared__ float sC[GROUPS_PER_BLOCK][4][16];   // local valid counts
    __shared__ float sA[GROUPS_PER_BLOCK][4][16];   // "no pad in sub-seg" flags

    const int tid  = threadIdx.x;
    const int wv   = tid >> 5;
    const int lane = tid & 31;
    const int sub  = lane & 15;        // row within 16-row group
    const int half = lane >> 4;        // 0 -> K=0,1   1 -> K=2,3
    const int g    = wv >> 1;          // row group within block
    const int s    = wv & 1;           // segment (positions [s*256, s*256+256))
    const int row  = blockIdx.x * ROWS_PER_BLOCK + g * 16 + sub;
    const int subseg = 2 * s + half;   // 0..3, 128 positions each
    const bool isLow = (half == 0);

    const float* rp = inp + (size_t)row * (LL * 3) + (size_t)subseg * (SUBSEG * 3);

    v8f aLR = {}, aUR = {}, aLF = {}, aUF = {};      // low/up x real/fake
    v2f bOnes; bOnes.x = 1.0f; bOnes.y = 1.0f;       // B = ones (layout-free)
    int alive = 1;                                   // intra-lane prefix validity
    int cnt = 0;

    for (int i = 0; i < NIT; ++i) {
        const float* p = rp + i * 6;                 // 2 consecutive (st,_,uid)
        float  st0 = p[0];
        float2 mid = *(const float2*)(p + 2);        // {uid0, st1} (8B aligned)
        float  uid0 = mid.x, st1 = mid.y;
        float  uid1 = p[5];

        int p0 = (st0 == PADV);
        int p1 = (st1 == PADV);
        int v0 = alive && !p0;                       // cumprod includes self
        int v1 = v0 && !p1;
        alive = v1;
        cnt += v0 + v1;

        // uid column is valid everywhere (pads only rewrite stance):
        // gather unconditionally, mask afterwards -> no exec-branches.
        float uw0 = w[(int)uid0];
        float uw1 = w[(int)uid1];

        float r0 = (v0 && st0 == 0.0f) ? uw0 : 0.0f;
        float f0 = (v0 && st0 != 0.0f) ? uw0 : 0.0f;
        float r1 = (v1 && st1 == 0.0f) ? uw1 : 0.0f;
        float f1 = (v1 && st1 != 0.0f) ? uw1 : 0.0f;

        // Separate lower-lane (K=0,1) and upper-lane (K=2,3) contributions
        // so the epilogue can apply the prefix-alive correction per sub-seg.
        v2f aRlo, aRup, aFlo, aFup;
        aRlo.x = isLow ? r0 : 0.0f;  aRlo.y = isLow ? r1 : 0.0f;
        aRup.x = isLow ? 0.0f : r0;  aRup.y = isLow ? 0.0f : r1;
        aFlo.x = isLow ? f0 : 0.0f;  aFlo.y = isLow ? f1 : 0.0f;
        aFup.x = isLow ? 0.0f : f0;  aFup.y = isLow ? 0.0f : f1;

        aLR = __builtin_amdgcn_wmma_f32_16x16x4_f32(false, aRlo, false, bOnes, (short)0, aLR, false, false);
        aUR = __builtin_amdgcn_wmma_f32_16x16x4_f32(false, aRup, false, bOnes, (short)0, aUR, false, false);
        aLF = __builtin_amdgcn_wmma_f32_16x16x4_f32(false, aFlo, false, bOnes, (short)0, aLF, false, false);
        aUF = __builtin_amdgcn_wmma_f32_16x16x4_f32(false, aFup, false, bOnes, (short)0, aUF, false, false);
    }

    // D layout: lane 0 holds rows 0..7 (all columns identical), lane 16 rows 8..15
    if (sub == 0) {
        #pragma unroll
        for (int i = 0; i < 8; ++i) {
            int r8 = half * 8 + i;
            sR[g][2 * s + 0][r8] = aLR[i];   // lower-lane stream = sub-seg 2s
            sR[g][2 * s + 1][r8] = aUR[i];   // upper-lane stream = sub-seg 2s+1
            sF[g][2 * s + 0][r8] = aLF[i];
            sF[g][2 * s + 1][r8] = aUF[i];
        }
    }
    sC[g][subseg][sub] = (float)cnt;
    sA[g][subseg][sub] = alive ? 1.0f : 0.0f;
    __syncthreads();

    if (tid < ROWS_PER_BLOCK) {
        const int gg = tid >> 4, r = tid & 15;
        float a0 = sA[gg][0][r], a1 = sA[gg][1][r], a2 = sA[gg][2][r];
        float m1 = a0, m2 = a0 * a1, m3 = m2 * a2;

        float real_pro = sR[gg][0][r] + m1 * sR[gg][1][r] + m2 * sR[gg][2][r] + m3 * sR[gg][3][r];
        float fake_pro = sF[gg][0][r] + m1 * sF[gg][1][r] + m2 * sF[gg][2][r] + m3 * sF[gg][3][r];
        float rn       = sC[gg][0][r] + m1 * sC[gg][1][r] + m2 * sC[gg][2][r] + m3 * sC[gg][3][r];

        float mx  = fmaxf(real_pro, fake_pro);
        float e0  = expf(real_pro - mx);
        float e1  = expf(fake_pro - mx);
        float inv = 1.0f / (e0 + e1);
        float pre0 = e0 * inv, pre1 = e1 * inv;

        float th0 = pre0 * rn, th1 = pre1 * rn;
        float a = th1, b = th0;
        float ssum = a + b;
        float mean = a / ssum;
        float var  = a * b / (ssum * ssum * (ssum + 1.0f));

        int orow = blockIdx.x * ROWS_PER_BLOCK + tid;
        out[orow * 2 + 0] = pre0;                  // user_pre
        out[orow * 2 + 1] = pre1;
        out[2 * BB + orow * 2 + 0] = mean;         // user_distribution
        out[2 * BB + orow * 2 + 1] = sqrtf(var);
        out[4 * BB + orow * 2 + 0] = th0;          // user_theta
        out[4 * BB + orow * 2 + 1] = th1;
    }
}

extern "C" void kernel_launch(void* const* d_in, const int* in_sizes, int n_in,
                              void* d_out, int out_size, void* d_ws, size_t ws_size,
                              hipStream_t stream)
{
    const float* inp = (const float*)d_in[0];   // (B, L, 3) f32
    const float* w   = (const float*)d_in[1];   // (N_USERS,) f32
    float* out = (float*)d_out;                 // user_pre | user_distribution | user_theta

    dim3 grid(BB / ROWS_PER_BLOCK);             // 512 blocks
    dim3 block(THREADS);                        // 128 threads = 4 waves
    hipLaunchKernelGGL(crowd_beta_kernel, grid, block, 0, stream, inp, w, out);
}